// KernelLearner_53060025975260
// MI455X (gfx1250) — compile-verified
//
#include <hip/hip_runtime.h>
#include <hip/hip_bf16.h>

// ---------------------------------------------------------------------------
// Sparse multi-order dispersion PSF applied via DFT-as-GEMM on the CDNA5
// tensor path.
//
//   y[b] = (1/N^2) * Re( conj(F) * ( sum_c (F x[b,c] F) .* (F k[c] F) ) * conj(F) )
//
// with F the symmetric 512x512 DFT matrix, complex data kept as planar
// real/imag fp32 planes so every transform stage is a real 512^3 GEMM
// executed with V_WMMA_F32_16X16X4_F32 (exact fp32 tensor math).
// B panels are staged into LDS by the Tensor Data Mover (double-buffered, so
// the DMA of panel i+1 overlaps the WMMAs of panel i) and consumed via
// ds_load; A fragments stream from global (b64 per lane, L1-resident).
//
// d_ws layout (floats, N = 512*512 = 262144):
//   Fr      @ 0        Fi      @ 1N
//   kdense  @ 2N..34N  (32 channel planes)
//   Kfr     @ 34N..66N Kfi @ 66N..98N
//   Tr @ 98N  Ti @ 99N  Ar @ 100N  Ai @ 101N  Yfr @ 102N  Yfi @ 103N
// total: 104N floats = ~109 MB of workspace.
// ---------------------------------------------------------------------------

#define NFFT 512
#define NPIX (NFFT * NFFT)
#define NCH  32
#define NB   8
#define NLOC 100000

// LDS B panel: 64 K-rows x 64 N-cols, row stride padded to 80 floats so the
// paired row reads (lanes 0-15 -> row k, lanes 16-31 -> row k+2) land on
// disjoint bank ranges (offset 2*80 mod 64 = 32 banks apart).
#define BP_ROW 80
#define KPAN   64

typedef float v2f __attribute__((ext_vector_type(2)));
typedef float v8f __attribute__((ext_vector_type(8)));
typedef unsigned int u32x4 __attribute__((ext_vector_type(4)));
typedef int i32x4 __attribute__((ext_vector_type(4)));
typedef int i32x8 __attribute__((ext_vector_type(8)));

// ---------------------------------------------------------------- elementwise
__global__ void zero_kernel(float* __restrict__ p, int n) {
    int i = blockIdx.x * blockDim.x + threadIdx.x;
    int stride = gridDim.x * blockDim.x;
    for (; i < n; i += stride) p[i] = 0.0f;
}

// Twiddle planes: F[m,n] = exp(-2*pi*i*m*n/512). Reduce m*n mod 512 exactly
// before the trig so large products stay precise.
__global__ void twiddle_kernel(float* __restrict__ Fr, float* __restrict__ Fi) {
    int i = blockIdx.x * blockDim.x + threadIdx.x;
    if (i >= NPIX) return;
    int m = i >> 9;
    int n = i & (NFFT - 1);
    int t = (m * n) & (NFFT - 1);
    float ang = -6.283185307179586f * (float)t / (float)NFFT;
    float s, c;
    sincosf(ang, &s, &c);
    Fr[i] = c;
    Fi[i] = s;
}

// Scatter ReLU'd trainable taps into the dense 32x512x512 kernel volume.
__global__ void scatter_kernel(const float* __restrict__ vk,
                               const int* __restrict__ idx,
                               float* __restrict__ kdense, int n) {
    int i = blockIdx.x * blockDim.x + threadIdx.x;
    if (i >= n) return;
    float v = vk[i];
    kdense[idx[i]] = v > 0.0f ? v : 0.0f;
}

// Yf += Xf * Kf  (complex, planar), accumulating the channel sum.
__global__ void cmul_acc_kernel(const float* __restrict__ ar, const float* __restrict__ ai,
                                const float* __restrict__ br, const float* __restrict__ bi,
                                float* __restrict__ yr, float* __restrict__ yi, int n) {
    int i = blockIdx.x * blockDim.x + threadIdx.x;
    int stride = gridDim.x * blockDim.x;
    for (; i < n; i += stride) {
        float xr = ar[i], xi = ai[i], kr = br[i], ki = bi[i];
        yr[i] += xr * kr - xi * ki;
        yi[i] += xr * ki + xi * kr;
    }
}

// ----------------------------------------------------------------- TDM stage
// Issue (no wait) a Tensor Data Mover load of B[kp:kp+64, n0:n0+64] (fp32,
// row-major, tensor 512x512) into LDS at lds_off with rows padded
// 64 DW -> +16 DW (BP_ROW = 80 floats). One wave issues the whole tile;
// TENSORcnt tracks completion (in-order per wave -> wait tensorcnt<=1 retires
// the older of two outstanding transfers).
// This toolchain exposes the 6-argument builtin:
//   (u32x4 g0, i32x8 g1, i32x4 g2, i32x4 g3, i32x8 g4, i32 cpol)
__device__ __forceinline__ void tdm_issue_b(const float* gsrc, unsigned lds_off) {
    unsigned long long ga = (unsigned long long)(uintptr_t)gsrc;
    u32x4 g0;
    g0[0] = 1u;                                   // count=1 user descriptor
    g0[1] = lds_off;                              // lds_addr (bytes)
    g0[2] = (unsigned)(ga & 0xFFFFFFFFu);         // global_addr[31:0]
    g0[3] = (unsigned)((ga >> 32) & 0x01FFFFFFu)  // global_addr[56:32]
          | (2u << 30);                           // type = 2 ("image")
    i32x8 g1;
    g1[0] = (int)((2u << 16)        // data_size = 4 bytes
                | (1u << 20)        // pad_enable
                | (5u << 22)        // pad_interval: 64 DWORDs
                | (15u << 25));     // pad_amount: 16 DWORDs
    g1[1] = (int)(512u << 16);      // tensor_dim0 = 512 (bits 79:48, low half)
    g1[2] = (int)(512u << 16);      // tensor_dim0 hi=0 | tensor_dim1 = 512 (low)
    g1[3] = (int)(64u << 16);       // tensor_dim1 hi=0 | tile_dim0 = 64
    g1[4] = 64;                     // tile_dim1 = 64, tile_dim2 = 0 (2D tile)
    g1[5] = 512;                    // tensor_dim0_stride = 512 elements
    g1[6] = 0;                      // stride0 hi | tensor_dim1_stride lo (unused, 2D)
    g1[7] = 0;
    i32x4 gz4 = {0, 0, 0, 0};       // groups 2/3 unused for 2D tiles
    i32x8 gz8 = {0, 0, 0, 0, 0, 0, 0, 0};
    __builtin_amdgcn_tensor_load_to_lds(g0, g1, gz4, gz4, gz8, 0);
}

// ------------------------------------------------------------------ WMMA GEMM
// C = alpha * A*B + beta * C for row-major 512x512 fp32 matrices.
// Block = 128 threads = 4 waves; block macro-tile 64(M) x 64(N); each wave
// owns a 16x64 strip (4 accumulators reuse one A fragment). K is processed in
// 64-wide panels: TDM double-buffers B panels through LDS (DMA of panel i+1
// overlaps compute of panel i), inner loop steps K by 4 with
// V_WMMA_F32_16X16X4_F32.
//
// Fragment layouts per CDNA5 ISA 7.12.2:
//   A 16x4 : lane L<16 -> row m0+L, K = {k0,k0+1}; lane>=16 -> K = {k0+2,k0+3}
//   B 4x16 : lane L<16 -> col n0+L, K = {k0,k0+1}; lane>=16 -> K = {k0+2,k0+3}
//   C 16x16: VGPR r -> row m0 + 8*(L>=16) + r, col n0 + (L&15)
__global__ __launch_bounds__(128)
void wmma_gemm_512(const float* __restrict__ A, const float* __restrict__ B,
                   float* __restrict__ C, float alpha, float beta) {
    __shared__ float Bs[2][KPAN * BP_ROW];

    const int lane = threadIdx.x & 31;
    const int wave = threadIdx.x >> 5;
    const int m0   = blockIdx.x * 64 + wave * 16;
    const int n0   = blockIdx.y * 64;
    const int half = lane >> 4;   // 0: K pair {0,1}; 1: K pair {2,3}
    const int l16  = lane & 15;

    v8f acc[4];
    acc[0] = v8f{}; acc[1] = v8f{}; acc[2] = v8f{}; acc[3] = v8f{};

    const float* Arow = A + (long)(m0 + l16) * NFFT + 2 * half; // b64 per step
    const unsigned bs_off[2] = { (unsigned)(uintptr_t)(&Bs[0][0]),
                                 (unsigned)(uintptr_t)(&Bs[1][0]) };

    // Prologue: kick off the DMA for panel 0.
    if (wave == 0)
        tdm_issue_b(B + n0, bs_off[0]);

    int buf = 0;
    for (int kp = 0; kp < NFFT; kp += KPAN, buf ^= 1) {
        if (wave == 0) {
            if (kp + KPAN < NFFT) {
                // Issue next panel into the other buffer, then retire the
                // older (current) transfer: TENSORcnt <= 1.
                tdm_issue_b(B + (long)(kp + KPAN) * NFFT + n0, bs_off[buf ^ 1]);
                __builtin_amdgcn_s_wait_tensorcnt(1);
            } else {
                __builtin_amdgcn_s_wait_tensorcnt(0);
            }
        }
        __syncthreads();

        // Prefetch the next A panel while this one computes.
        __builtin_prefetch(Arow + kp + KPAN, 0, 3);

        const float* Bb = &Bs[buf][0];
        #pragma unroll 4
        for (int k = 0; k < KPAN; k += 4) {
            v2f a = *(const v2f*)(Arow + kp + k);
            const float* Br = Bb + (k + 2 * half) * BP_ROW + l16;

            v2f b0 = { Br[0],  Br[BP_ROW + 0]  };
            v2f b1 = { Br[16], Br[BP_ROW + 16] };
            v2f b2 = { Br[32], Br[BP_ROW + 32] };
            v2f b3 = { Br[48], Br[BP_ROW + 48] };

            acc[0] = __builtin_amdgcn_wmma_f32_16x16x4_f32(false, a, false, b0,
                                                           (short)0, acc[0], false, false);
            acc[1] = __builtin_amdgcn_wmma_f32_16x16x4_f32(false, a, false, b1,
                                                           (short)0, acc[1], false, false);
            acc[2] = __builtin_amdgcn_wmma_f32_16x16x4_f32(false, a, false, b2,
                                                           (short)0, acc[2], false, false);
            acc[3] = __builtin_amdgcn_wmma_f32_16x16x4_f32(false, a, false, b3,
                                                           (short)0, acc[3], false, false);
        }
        __syncthreads();
    }

    #pragma unroll
    for (int j = 0; j < 4; ++j) {
        #pragma unroll
        for (int r = 0; r < 8; ++r) {
            long idx = (long)(m0 + 8 * half + r) * NFFT + n0 + 16 * j + l16;
            float v = alpha * acc[j][r];
            if (beta != 0.0f) v += beta * C[idx];
            C[idx] = v;
        }
    }
}

// ------------------------------------------------------------------ host side
static inline void gemm(const float* A, const float* B, float* C,
                        float alpha, float beta, hipStream_t s) {
    dim3 grid(NFFT / 64, NFFT / 64);   // 8 x 8 blocks, 128 threads each
    wmma_gemm_512<<<grid, 128, 0, s>>>(A, B, C, alpha, beta);
}

// Forward 2D DFT of real plane X -> complex planes (Zr, Zi), planar algebra:
//   T = X*F  (Tr = X*Fr, Ti = X*Fi)
//   Z = F*T  (Zr = Fr*Tr - Fi*Ti, Zi = Fr*Ti + Fi*Tr)
static inline void fwd_dft(const float* X, float* Zr, float* Zi,
                           float* Tr, float* Ti,
                           const float* Fr, const float* Fi, hipStream_t s) {
    gemm(X,  Fr, Tr,  1.0f, 0.0f, s);
    gemm(X,  Fi, Ti,  1.0f, 0.0f, s);
    gemm(Fr, Tr, Zr,  1.0f, 0.0f, s);
    gemm(Fi, Ti, Zr, -1.0f, 1.0f, s);
    gemm(Fr, Ti, Zi,  1.0f, 0.0f, s);
    gemm(Fi, Tr, Zi,  1.0f, 1.0f, s);
}

extern "C" void kernel_launch(void* const* d_in, const int* in_sizes, int n_in,
                              void* d_out, int out_size, void* d_ws, size_t ws_size,
                              hipStream_t stream) {
    const float* x   = (const float*)d_in[0];  // (8, 32, 512, 512) f32
    // d_in[1] is the all-zero kernel template (unused).
    const float* vk  = (const float*)d_in[2];  // (100000,) f32
    const int*   loc = (const int*)d_in[3];    // (100000,) int

    float* ws  = (float*)d_ws;
    float* out = (float*)d_out;

    const long N = NPIX;
    float* Fr     = ws;
    float* Fi     = ws + 1 * N;
    float* kdense = ws + 2 * N;           // 32 planes
    float* Kfr    = ws + 34 * N;          // 32 planes
    float* Kfi    = ws + 66 * N;          // 32 planes
    float* Tr     = ws + 98 * N;
    float* Ti     = ws + 99 * N;
    float* Ar     = ws + 100 * N;
    float* Ai     = ws + 101 * N;
    float* Yfr    = ws + 102 * N;
    float* Yfi    = ws + 103 * N;
    (void)ws_size; (void)in_sizes; (void)n_in; (void)out_size;

    // 1) Build the dense sparse-PSF volume: zero + scatter(ReLU(vk)).
    zero_kernel<<<2048, 256, 0, stream>>>(kdense, NCH * NPIX);
    twiddle_kernel<<<(NPIX + 255) / 256, 256, 0, stream>>>(Fr, Fi);
    scatter_kernel<<<(NLOC + 255) / 256, 256, 0, stream>>>(vk, loc, kdense, NLOC);

    // 2) Forward transform every kernel channel: Kf[c] = F * k[c] * F.
    for (int c = 0; c < NCH; ++c)
        fwd_dft(kdense + (long)c * N, Kfr + (long)c * N, Kfi + (long)c * N,
                Tr, Ti, Fr, Fi, stream);

    // 3) Per batch: transform channels, accumulate Yf = sum_c Xf.*Kf,
    //    then inverse transform (real part, 1/N^2 folded into alpha).
    const float inv_n2 = 1.0f / ((float)NFFT * (float)NFFT);
    for (int b = 0; b < NB; ++b) {
        zero_kernel<<<1024, 256, 0, stream>>>(Yfr, NPIX);
        zero_kernel<<<1024, 256, 0, stream>>>(Yfi, NPIX);

        for (int c = 0; c < NCH; ++c) {
            const float* xp = x + ((long)b * NCH + c) * N;
            fwd_dft(xp, Ar, Ai, Tr, Ti, Fr, Fi, stream);
            cmul_acc_kernel<<<1024, 256, 0, stream>>>(
                Ar, Ai, Kfr + (long)c * N, Kfi + (long)c * N, Yfr, Yfi, NPIX);
        }

        // Inverse with G = conj(F) = Fr - i*Fi:
        //   T = G*Yf : Tr = Fr*Yfr + Fi*Yfi ; Ti = Fr*Yfi - Fi*Yfr
        gemm(Fr, Yfr, Tr,  1.0f, 0.0f, stream);
        gemm(Fi, Yfi, Tr,  1.0f, 1.0f, stream);
        gemm(Fr, Yfi, Ti,  1.0f, 0.0f, stream);
        gemm(Fi, Yfr, Ti, -1.0f, 1.0f, stream);
        //   y = Re(T*G)/N^2 = (Tr*Fr + Ti*Fi)/N^2
        float* yb = out + (long)b * N;
        gemm(Tr, Fr, yb, inv_n2, 0.0f, stream);
        gemm(Ti, Fi, yb, inv_n2, 1.0f, stream);
    }
}